// KNNClusteringModule_317827580064
// MI455X (gfx1250) — compile-verified
//
#include <hip/hip_runtime.h>
#include <cstdint>
#include <math.h>

typedef __attribute__((ext_vector_type(2))) float v2f;
typedef __attribute__((ext_vector_type(8))) float v8f;

#define NPTS 8192
#define DIM  512
#define KC   8
#define NN   5

// d_out flat layout (reference return order)
#define ENC_OFF    0            // 8192*32
#define ASSIGN_OFF 262144       // 8192*8
#define KNN_OFF    327680       // 8192*5
#define STATS_OFF  368640       // 8192*3
#define SC_OFF     393216       // loss, intra, inter

__device__ __forceinline__ v8f wmma4(v2f a, v2f b, v8f c) {
  return __builtin_amdgcn_wmma_f32_16x16x4_f32(false, a, false, b, (short)0, c, false, false);
}

__device__ __forceinline__ float wred_sum(float v) {
#pragma unroll
  for (int m = 1; m < 32; m <<= 1) v += __shfl_xor(v, m, 32);
  return v;
}
__device__ __forceinline__ float wred_max(float v) {
#pragma unroll
  for (int m = 1; m < 32; m <<= 1) v = fmaxf(v, __shfl_xor(v, m, 32));
  return v;
}

// insert into ascending 5-element (smallest-5) sorted list
__device__ __forceinline__ void ins5(float v, int id, float (&tv)[5], int (&ti)[5]) {
  if (v < tv[4]) {
    tv[4] = v; ti[4] = id;
#pragma unroll
    for (int s = 4; s > 0; --s) {
      if (tv[s] < tv[s - 1]) {
        float tf = tv[s]; tv[s] = tv[s - 1]; tv[s - 1] = tf;
        int   t2 = ti[s]; ti[s] = ti[s - 1]; ti[s - 1] = t2;
      }
    }
  }
}

// ---------------- Kernel 1: row sumsq + stats (mu, sd, entropy) ----------------
__global__ __launch_bounds__(256) void stats_sq_kernel(const float* __restrict__ x,
                                                       float* __restrict__ out,
                                                       float* __restrict__ sq) {
  const int lane = threadIdx.x & 31, w = threadIdx.x >> 5;
  const int row = blockIdx.x * 8 + w;
  float xv[16];
#pragma unroll
  for (int t = 0; t < 16; ++t) xv[t] = x[(size_t)row * DIM + lane + 32 * t];
  float s = 0.f, s2 = 0.f, mx = -3.0e38f;
#pragma unroll
  for (int t = 0; t < 16; ++t) { s += xv[t]; s2 += xv[t] * xv[t]; mx = fmaxf(mx, xv[t]); }
  s = wred_sum(s); s2 = wred_sum(s2); mx = wred_max(mx);
  float Z = 0.f, S = 0.f;
#pragma unroll
  for (int t = 0; t < 16; ++t) { float e = expf(xv[t] - mx); Z += e; S += xv[t] * e; }
  Z = wred_sum(Z); S = wred_sum(S);
  if (lane == 0) {
    float mu = s * (1.0f / DIM);
    float var = (s2 - s * s * (1.0f / DIM)) * (1.0f / (DIM - 1));
    float sd = sqrtf(fmaxf(var, 0.0f)) + 1e-8f;
    float ent = mx + logf(Z) - S / Z;
    out[STATS_OFF + row * 3 + 0] = mu;
    out[STATS_OFF + row * 3 + 1] = sd;
    out[STATS_OFF + row * 3 + 2] = ent;
    sq[row] = s2;
  }
}

// ---------------- Kernel 2: cluster distances, softmax assign, intra partials ----------------
__global__ __launch_bounds__(256) void assign_kernel(const float* __restrict__ x,
                                                     const float* __restrict__ cc,
                                                     const float* __restrict__ cw,
                                                     const float* __restrict__ temperature,
                                                     const float* __restrict__ sq,
                                                     float* __restrict__ out,
                                                     float* __restrict__ intra_part) {
  __shared__ float cent[KC * DIM];
  __shared__ float csq[KC];
  const int tid = threadIdx.x, lane = tid & 31, w = tid >> 5;
  for (int i = tid; i < KC * DIM; i += 256) cent[i] = cc[i];
  __syncthreads();
  {
    float p = 0.f;
#pragma unroll
    for (int t = 0; t < 16; ++t) { float v = cent[w * DIM + lane + 32 * t]; p += v * v; }
    p = wred_sum(p);
    if (lane == 0) csq[w] = p;
  }
  __syncthreads();

  const int row = blockIdx.x * 8 + w;
  float xv[16];
#pragma unroll
  for (int t = 0; t < 16; ++t) xv[t] = x[(size_t)row * DIM + lane + 32 * t];
  const float sqr = sq[row];
  const float temp = temperature[0];

  float dc[KC], aa[KC];
  float m = -3.0e38f;
#pragma unroll
  for (int k = 0; k < KC; ++k) {
    float p = 0.f;
#pragma unroll
    for (int t = 0; t < 16; ++t) p += xv[t] * cent[k * DIM + lane + 32 * t];
    float dot = wred_sum(p);
    float d2 = sqr + csq[k] - 2.0f * dot;
    dc[k] = (d2 > 0.0f) ? sqrtf(d2) : 0.0f;
    m = fmaxf(m, -dc[k] / temp);
  }
  float Zs = 0.f;
#pragma unroll
  for (int k = 0; k < KC; ++k) { aa[k] = expf(-dc[k] / temp - m); Zs += aa[k]; }
  float ip = 0.f;
#pragma unroll
  for (int k = 0; k < KC; ++k) { aa[k] = aa[k] / Zs * cw[k]; ip += dc[k] * aa[k]; }
  if (lane == 0) {
#pragma unroll
    for (int k = 0; k < KC; ++k) out[ASSIGN_OFF + row * KC + k] = aa[k];
    intra_part[row] = ip;
  }
}

// ---------------- Kernel 3: fused pd = cdist(x,x) via f32 WMMA + noisy top-5 ----------------
__global__ __launch_bounds__(256) void pd_topk_kernel(const float* __restrict__ xg,
                                                      const float* __restrict__ noise,
                                                      const float* __restrict__ sq,
                                                      int* __restrict__ knn_idx) {
  __shared__ float At[16 * 516];           // 16 rows x 512, padded stride for bank spread
  __shared__ float mv[8][16][5];
  __shared__ int   mi[8][16][5];
  const int tid = threadIdx.x;
  const int lane = tid & 31, w = tid >> 5;
  const int hl = lane >> 4, ln = lane & 15;
  const int mt = blockIdx.x;

  // Stage A-tile (16x512) into LDS with async global->LDS copies (gfx1250 path)
  const float* gsrc = xg + (size_t)mt * 16 * DIM;
#pragma unroll
  for (int it = 0; it < 8; ++it) {
    int f = tid * 4 + it * 1024;
    int r = f >> 9, c = f & 511;
    uint32_t loff = (uint32_t)(uintptr_t)(&At[r * 516 + c]);
    uint64_t gaddr = (uint64_t)(uintptr_t)(gsrc + f);
    asm volatile("global_load_async_to_lds_b128 %0, %1, off" :: "v"(loff), "v"(gaddr) : "memory");
  }
  asm volatile("s_wait_asynccnt 0" ::: "memory");
  __syncthreads();

  float sqrow[8];
#pragma unroll
  for (int v = 0; v < 8; ++v) sqrow[v] = sq[mt * 16 + v + 8 * hl];

  float tv[8][5]; int ti[8][5];
#pragma unroll
  for (int v = 0; v < 8; ++v)
#pragma unroll
    for (int s = 0; s < 5; ++s) { tv[v][s] = 3.0e38f; ti[v][s] = 0; }

  const float* arow = &At[ln * 516 + 2 * hl];
  for (int it = 0; it < 64; ++it) {
    const int jt = it * 8 + w;
    const int j = jt * 16 + ln;
    const float sqj = sq[j];
    const float* brow = xg + (size_t)j * DIM + 2 * hl;

    // Issue the HBM-bound noise stream FIRST so the whole GEMM hides its latency.
    float nz[8];
#pragma unroll
    for (int v = 0; v < 8; ++v) {
      const int i = mt * 16 + v + 8 * hl;
      nz[v] = __builtin_nontemporal_load(noise + (size_t)i * NPTS + j);
    }
    if (it + 1 < 64)  // look-ahead prefetch of next noise stripe (global_prefetch_b8)
      __builtin_prefetch(noise + (size_t)(mt * 16 + 8 * hl) * NPTS + j + 128, 0, 0);

    // Software-pipelined dual-chain WMMA GEMM: fetch chunk k+1 while chunk k computes.
    v8f c0 = {}; v8f c1 = {};
    v2f aR[4], bR[4];
#pragma unroll
    for (int u = 0; u < 4; ++u) {
      aR[u] = *(const v2f*)(arow + u * 4);
      bR[u] = *(const v2f*)(brow + u * 4);
    }
#pragma unroll
    for (int ch = 0; ch < 32; ++ch) {
      v2f aN[4], bN[4];
      if (ch < 31) {
        const int base = (ch + 1) * 16;
#pragma unroll
        for (int u = 0; u < 4; ++u) {
          aN[u] = *(const v2f*)(arow + base + u * 4);
          bN[u] = *(const v2f*)(brow + base + u * 4);
        }
      }
      c0 = wmma4(aR[0], bR[0], c0);
      c1 = wmma4(aR[1], bR[1], c1);
      c0 = wmma4(aR[2], bR[2], c0);
      c1 = wmma4(aR[3], bR[3], c1);
      if (ch < 31) {
#pragma unroll
        for (int u = 0; u < 4; ++u) { aR[u] = aN[u]; bR[u] = bN[u]; }
      }
    }

#pragma unroll
    for (int v = 0; v < 8; ++v) {
      const int i = mt * 16 + v + 8 * hl;
      float dot = c0[v] + c1[v];
      float d2 = sqrow[v] + sqj - 2.0f * dot;
      float pd = (d2 > 0.0f) ? sqrtf(d2) : 0.0f;
      float noisy = pd + nz[v] * 1e-6f;
      if (i != j) ins5(noisy, j, tv[v], ti[v]);
    }
  }

  // merge the 16 column-groups of this wave via lane shuffles
#pragma unroll
  for (int m = 1; m < 16; m <<= 1) {
#pragma unroll
    for (int v = 0; v < 8; ++v) {
      float ov[5]; int oi[5];
#pragma unroll
      for (int s = 0; s < 5; ++s) {
        ov[s] = __shfl_xor(tv[v][s], m, 32);
        oi[s] = __shfl_xor(ti[v][s], m, 32);
      }
#pragma unroll
      for (int s = 0; s < 5; ++s) ins5(ov[s], oi[s], tv[v], ti[v]);
    }
  }
  if (ln == 0) {
#pragma unroll
    for (int v = 0; v < 8; ++v)
#pragma unroll
      for (int s = 0; s < 5; ++s) { mv[w][v + 8 * hl][s] = tv[v][s]; mi[w][v + 8 * hl][s] = ti[v][s]; }
  }
  __syncthreads();
  if (w == 0 && lane < 16) {
    float fv[5]; int fi[5];
#pragma unroll
    for (int s = 0; s < 5; ++s) { fv[s] = 3.0e38f; fi[s] = 0; }
    for (int ww = 0; ww < 8; ++ww)
#pragma unroll
      for (int s = 0; s < 5; ++s) ins5(mv[ww][lane][s], mi[ww][lane][s], fv, fi);
#pragma unroll
    for (int s = 0; s < 5; ++s) knn_idx[(mt * 16 + lane) * 5 + s] = fi[s];
  }
}

// ---------------- Kernel 4: clean knn distances from selected indices ----------------
__global__ __launch_bounds__(256) void knn_finalize_kernel(const float* __restrict__ x,
                                                           const float* __restrict__ sq,
                                                           const int* __restrict__ knn_idx,
                                                           float* __restrict__ out) {
  const int lane = threadIdx.x & 31, w = threadIdx.x >> 5;
  const int i = blockIdx.x * 8 + w;
  float xv[16];
#pragma unroll
  for (int t = 0; t < 16; ++t) xv[t] = x[(size_t)i * DIM + lane + 32 * t];
  const float sqi = sq[i];
#pragma unroll
  for (int s = 0; s < NN; ++s) {
    int j = knn_idx[i * 5 + s];
    float p = 0.f;
#pragma unroll
    for (int t = 0; t < 16; ++t) p += xv[t] * x[(size_t)j * DIM + lane + 32 * t];
    float dot = wred_sum(p);
    if (lane == 0) {
      float d2 = sqi + sq[j] - 2.0f * dot;
      out[KNN_OFF + i * 5 + s] = (d2 > 0.0f) ? sqrtf(d2) : 0.0f;
    }
  }
}

// ---------------- Kernel 5: MLP encoder with f32 WMMA (16->64 relu ->32) ----------------
__global__ __launch_bounds__(256) void mlp_kernel(const float* __restrict__ w1,
                                                  const float* __restrict__ b1,
                                                  const float* __restrict__ w2,
                                                  const float* __restrict__ b2,
                                                  float* out) {
  __shared__ float fs[8][16 * 20];   // feats 16x16, padded
  __shared__ float hs[8][16 * 68];   // hidden 16x64, padded
  const int lane = threadIdx.x & 31, w = threadIdx.x >> 5;
  const int hl = lane >> 4, ln = lane & 15;
  const int tile = blockIdx.x * 8 + w;

  float* fsw = &fs[w][0];
  for (int idx = lane; idx < 256; idx += 32) {
    int r = idx >> 4, c = idx & 15;
    int row = tile * 16 + r;
    float v;
    if (c < 8)       v = out[ASSIGN_OFF + row * 8 + c];
    else if (c < 13) v = out[KNN_OFF + row * 5 + (c - 8)];
    else             v = out[STATS_OFF + row * 3 + (c - 13)];
    fsw[r * 20 + c] = v;
  }
  __syncthreads();

  float* hsw = &hs[w][0];
#pragma unroll
  for (int nt = 0; nt < 4; ++nt) {
    const int col = nt * 16 + ln;
    const float bb = b1[col];
    v8f c;
#pragma unroll
    for (int q = 0; q < 8; ++q) c[q] = bb;
#pragma unroll
    for (int k0 = 0; k0 < 16; k0 += 4) {
      int k = k0 + 2 * hl;
      v2f a = *(const v2f*)(fsw + ln * 20 + k);
      v2f b; b.x = w1[k * 64 + col]; b.y = w1[(k + 1) * 64 + col];
      c = wmma4(a, b, c);
    }
#pragma unroll
    for (int v = 0; v < 8; ++v) hsw[(v + 8 * hl) * 68 + col] = fmaxf(c[v], 0.0f);
  }
  __syncthreads();

#pragma unroll
  for (int nt = 0; nt < 2; ++nt) {
    const int col = nt * 16 + ln;
    const float bb = b2[col];
    v8f c;
#pragma unroll
    for (int q = 0; q < 8; ++q) c[q] = bb;
#pragma unroll
    for (int k0 = 0; k0 < 64; k0 += 4) {
      int k = k0 + 2 * hl;
      v2f a = *(const v2f*)(hsw + ln * 68 + k);
      v2f b; b.x = w2[k * 32 + col]; b.y = w2[(k + 1) * 32 + col];
      c = wmma4(a, b, c);
    }
#pragma unroll
    for (int v = 0; v < 8; ++v)
      out[ENC_OFF + (size_t)(tile * 16 + v + 8 * hl) * 32 + col] = c[v];
  }
}

// ---------------- Kernel 6: inter-center distance, intra reduction, loss ----------------
__global__ __launch_bounds__(64) void scalars_kernel(const float* __restrict__ cc,
                                                     const float* __restrict__ intra_part,
                                                     float* __restrict__ out) {
  __shared__ float sm[64];
  __shared__ float si[64];
  const int t = threadIdx.x;
  const int a = t >> 3, b = t & 7;
  float da = 0.f, db = 0.f, dab = 0.f;
  for (int d = 0; d < DIM; ++d) {
    float va = cc[a * DIM + d], vb = cc[b * DIM + d];
    da += va * va; db += vb * vb; dab += va * vb;
  }
  float d2 = da + db - 2.0f * dab;
  float pd = (d2 > 0.0f) ? sqrtf(d2) : 0.0f;
  sm[t] = (a != b) ? pd : 0.0f;

  float ps = 0.f;
  for (int i = t; i < NPTS; i += 64) ps += intra_part[i];
  si[t] = ps;
  __syncthreads();
  if (t == 0) {
    float inter = 0.f, isum = 0.f;
    for (int i = 0; i < 64; ++i) { inter += sm[i]; isum += si[i]; }
    inter /= 56.0f;                              // K*K - K off-diagonal entries
    float intra = isum / (float)(NPTS * KC);
    out[SC_OFF + 0] = intra - 0.1f * inter;      // loss
    out[SC_OFF + 1] = intra;
    out[SC_OFF + 2] = inter;
  }
}

extern "C" void kernel_launch(void* const* d_in, const int* in_sizes, int n_in,
                              void* d_out, int out_size, void* d_ws, size_t ws_size,
                              hipStream_t stream) {
  const float* x     = (const float*)d_in[0];
  const float* noise = (const float*)d_in[1];
  const float* cc    = (const float*)d_in[2];
  const float* w1    = (const float*)d_in[3];
  const float* b1    = (const float*)d_in[4];
  const float* w2    = (const float*)d_in[5];
  const float* b2    = (const float*)d_in[6];
  const float* temp  = (const float*)d_in[7];
  const float* cw    = (const float*)d_in[8];
  float* out = (float*)d_out;

  float* wsf        = (float*)d_ws;
  float* sq         = wsf;               // [8192]
  float* intra_part = wsf + NPTS;        // [8192]
  int*   knn_idx    = (int*)(wsf + 2 * NPTS); // [8192*5]

  stats_sq_kernel<<<NPTS / 8, 256, 0, stream>>>(x, out, sq);
  assign_kernel<<<NPTS / 8, 256, 0, stream>>>(x, cc, cw, temp, sq, out, intra_part);
  pd_topk_kernel<<<NPTS / 16, 256, 0, stream>>>(x, noise, sq, knn_idx);
  knn_finalize_kernel<<<NPTS / 8, 256, 0, stream>>>(x, sq, knn_idx, out);
  mlp_kernel<<<NPTS / 16 / 8, 256, 0, stream>>>(w1, b1, w2, b2, out);
  scalars_kernel<<<1, 64, 0, stream>>>(cc, intra_part, out);
}